// ANodeDecoder_7705171329027
// MI455X (gfx1250) — compile-verified
//
#include <hip/hip_runtime.h>
#include <hip/hip_bf16.h>
#include <math.h>

// Problem constants (from reference)
#define NN    2048
#define BB    8
#define LL    64
#define DD    16      // D_OUT (head dim)
#define HH    4       // heads
#define HB    (HH*BB) // 32
#define ADA   16
#define KC    64      // K-chunk staged in LDS for the big GEMM

typedef __attribute__((ext_vector_type(2))) float v2f;
typedef __attribute__((ext_vector_type(8))) float v8f;
typedef __attribute__((ext_vector_type(4))) unsigned int u32x4;
typedef __attribute__((ext_vector_type(4))) int i32x4;
typedef __attribute__((ext_vector_type(8))) int i32x8;

#if defined(__has_builtin)
# if __has_builtin(__builtin_amdgcn_tensor_load_to_lds)
#  define HAVE_TDM 1
# endif
#endif
#ifndef HAVE_TDM
# define HAVE_TDM 0
#endif

// fp32 WMMA: D(16x16,f32) = A(16x4,f32) * B(4x16,f32) + C
// A-frag: lane l holds row (l&15), K = (l>>4)*2 + {0,1}
// B-frag: lane l holds col (l&15), K = (l>>4)*2 + {0,1}
// C/D:    lane l, vgpr r -> element (m = r + 8*(l>>4), n = l&15)
__device__ __forceinline__ v8f wmma_f32_k4(v2f a, v2f b, v8f c) {
  return __builtin_amdgcn_wmma_f32_16x16x4_f32(
      /*neg_a=*/false, a, /*neg_b=*/false, b,
      /*c_mod=*/(short)0, c, /*reuse_a=*/false, /*reuse_b=*/false);
}

#if HAVE_TDM
// Issue a TDM load of a 2D fp32 tile (rows x cols) into LDS.
// D# packing per CDNA5 ISA ch.8:
//  group0: [1:0]=count=1, [63:32]=lds_addr, [120:64]=global_addr, [127:126]=type(2)
//  group1: [17:16]=data_size(2=4B), [79:48]=tensor_dim0, [111:80]=tensor_dim1,
//          [127:112]=tile_dim0, [143:128]=tile_dim1, [207:160]=tensor_dim0_stride
__device__ __forceinline__ void tdm_load_tile_f32(unsigned int lds_addr,
                                                  const float* gsrc,
                                                  unsigned int rows,
                                                  unsigned int cols,
                                                  unsigned int row_stride_elems) {
  const unsigned long long ga = (unsigned long long)(uintptr_t)gsrc;
  u32x4 g0;
  g0.x = 1u;                                      // count=1 (valid user D#)
  g0.y = lds_addr;                                // LDS byte address of tile
  g0.z = (unsigned int)(ga & 0xffffffffu);        // global_addr[31:0]
  g0.w = (unsigned int)((ga >> 32) & 0x01ffffffu) // global_addr[56:32]
         | (2u << 30);                            // type = 2 ("image")
  i32x8 g1;
  g1[0] = (int)(2u << 16);                        // data_size = 4 bytes
  g1[1] = (int)((cols & 0xffffu) << 16);          // tensor_dim0[15:0]
  g1[2] = (int)(((cols >> 16) & 0xffffu)          // tensor_dim0[31:16]
                | ((rows & 0xffffu) << 16));      // tensor_dim1[15:0]
  g1[3] = (int)(((rows >> 16) & 0xffffu)          // tensor_dim1[31:16]
                | ((cols & 0xffffu) << 16));      // tile_dim0
  g1[4] = (int)(rows & 0xffffu);                  // tile_dim1 (tile_dim2=0)
  g1[5] = (int)row_stride_elems;                  // tensor_dim0_stride[31:0]
  g1[6] = 0;                                      // stride hi / dim1_stride lo
  g1[7] = 0;
  i32x4 z4 = {0, 0, 0, 0};
#if __clang_major__ >= 23
  i32x8 z8 = {0, 0, 0, 0, 0, 0, 0, 0};
  __builtin_amdgcn_tensor_load_to_lds(g0, g1, z4, z4, z8, 0);
#else
  __builtin_amdgcn_tensor_load_to_lds(g0, g1, z4, z4, 0);
#endif
}
#endif  // HAVE_TDM

// ---------------------------------------------------------------------------
// K1: A = softmax(relu(E @ E^T), axis=1)   E:[N,ADA]  A:[N,N]
// one block (256 threads) per row n; each thread owns 8 columns
// ---------------------------------------------------------------------------
__global__ void adj_kernel(const float* __restrict__ E, float* __restrict__ A) {
  const int n = blockIdx.x;
  const int t = threadIdx.x;
  __shared__ float En[ADA];
  __shared__ float red[256];

  if (t < ADA) En[t] = E[n * ADA + t];
  __syncthreads();

  float sv[8];
  float lmax = -INFINITY;
#pragma unroll
  for (int j = 0; j < 8; ++j) {
    const int m = t + j * 256;
    const float* Em = E + (size_t)m * ADA;
    float d = 0.f;
#pragma unroll
    for (int q = 0; q < ADA; ++q) d = fmaf(En[q], Em[q], d);
    d = fmaxf(d, 0.f);  // relu
    sv[j] = d;
    lmax = fmaxf(lmax, d);
  }

  red[t] = lmax;
  __syncthreads();
  for (int s = 128; s > 0; s >>= 1) {
    if (t < s) red[t] = fmaxf(red[t], red[t + s]);
    __syncthreads();
  }
  const float rmax = red[0];
  __syncthreads();

  float lsum = 0.f;
#pragma unroll
  for (int j = 0; j < 8; ++j) {
    sv[j] = __expf(sv[j] - rmax);
    lsum += sv[j];
  }
  red[t] = lsum;
  __syncthreads();
  for (int s = 128; s > 0; s >>= 1) {
    if (t < s) red[t] += red[t + s];
    __syncthreads();
  }
  const float inv = 1.f / red[0];

#pragma unroll
  for (int j = 0; j < 8; ++j)
    A[(size_t)n * NN + t + j * 256] = sv[j] * inv;
}

// ---------------------------------------------------------------------------
// K2: xg2[b] = A @ c[b]   ([N,N] @ [N,L]) — fp32 WMMA with a TDM-fed,
// double-buffered LDS pipeline: wave 0 issues the async tensor DMA for chunk
// i+1 right after publishing chunk i, so WMMA compute overlaps the DMA.
// grid: (N/64, B); block: 128 (4 waves). Each wave owns a 16x64 output strip.
// ---------------------------------------------------------------------------
__global__ void gemm_A_c(const float* __restrict__ A, const float* __restrict__ c,
                         float* __restrict__ out) {
  __shared__ float tile[2][KC * LL];  // 2 x 16 KB ping-pong buffers
  const int b    = blockIdx.y;
  const int wave = threadIdx.x >> 5;
  const int lane = threadIdx.x & 31;
  const int half = lane >> 4;
  const int lid  = lane & 15;
  const int m0   = blockIdx.x * 64 + wave * 16;

  const float* Arow = A + (size_t)(m0 + lid) * NN;  // this lane's A row
  const float* Cb   = c + (size_t)b * NN * LL;

  v8f acc[4] = {{}, {}, {}, {}};
  const int nchunks = NN / KC;

#if HAVE_TDM
  if (threadIdx.x < 32)  // prologue: DMA chunk 0 into buffer 0
    tdm_load_tile_f32((unsigned int)(uintptr_t)(&tile[0][0]), Cb, KC, LL, LL);
#endif

  for (int ci = 0; ci < nchunks; ++ci) {
    const int k0 = ci * KC;
#if HAVE_TDM
    if (threadIdx.x < 32)
      __builtin_amdgcn_s_wait_tensorcnt(0);   // chunk ci landed in LDS
    __syncthreads();                          // publish to all 4 waves
    if (ci + 1 < nchunks && threadIdx.x < 32) // overlap DMA of ci+1 w/ compute
      tdm_load_tile_f32((unsigned int)(uintptr_t)(&tile[(ci + 1) & 1][0]),
                        Cb + (size_t)(k0 + KC) * LL, KC, LL, LL);
#else
    __syncthreads();                          // previous buffer consumed
    for (int j = threadIdx.x; j < KC * LL; j += 128)
      tile[ci & 1][j] = Cb[(size_t)k0 * LL + j];
    __syncthreads();
#endif
    const float* tl = tile[ci & 1];

    __builtin_prefetch(Arow + k0 + KC, 0, 0); // next A chunk -> global_prefetch_b8

#pragma unroll 4
    for (int ks = 0; ks < KC; ks += 4) {
      const int kr = ks + half * 2;           // K row within tile
      const int ka = k0 + kr;
      v2f af = { Arow[ka], Arow[ka + 1] };    // one A-frag feeds 4 WMMAs
#pragma unroll
      for (int nt = 0; nt < 4; ++nt) {
        v2f bf = { tl[kr * LL + nt * 16 + lid],
                   tl[(kr + 1) * LL + nt * 16 + lid] };
        acc[nt] = wmma_f32_k4(af, bf, acc[nt]);
      }
    }
  }

  float* Ob = out + (size_t)b * NN * LL;
#pragma unroll
  for (int nt = 0; nt < 4; ++nt)
#pragma unroll
    for (int r = 0; r < 8; ++r)
      Ob[(size_t)(m0 + r + 8 * half) * LL + nt * 16 + lid] = acc[nt][r];
}

// ---------------------------------------------------------------------------
// K3: ct[b,n,o] = tanh( sum_{k,i,d} xg[b,k,n,i] E[n,d] Wp[d,k,i,o] + (E@bp)[n,o] )
// one block (256 threads) per node n; Wn = E[n]·Wp held in LDS (32 KB)
// xg[b,0] = c, xg[b,1] = xg2
// ---------------------------------------------------------------------------
__global__ void gcn_kernel(const float* __restrict__ c, const float* __restrict__ xg2,
                           const float* __restrict__ E, const float* __restrict__ Wp,
                           const float* __restrict__ bp, float* __restrict__ ct) {
  const int n = blockIdx.x;
  const int t = threadIdx.x;
  __shared__ float En[ADA];
  __shared__ float Wn[2 * LL * LL];   // [ki][o], 8192 floats = 32 KB
  __shared__ float xl[BB][2 * LL];    // [b][ki], 4 KB

  if (t < ADA) En[t] = E[n * ADA + t];
  __syncthreads();

  // Wn[ki*64+o] = sum_d En[d] * Wp[d*8192 + ki*64 + o]
  for (int j = t; j < 2 * LL * LL; j += 256) {
    float s = 0.f;
#pragma unroll
    for (int d = 0; d < ADA; ++d) s = fmaf(En[d], Wp[d * 8192 + j], s);
    Wn[j] = s;
  }

  // stage xg for this node
  for (int j = t; j < BB * 2 * LL; j += 256) {
    const int b  = j >> 7;          // /128
    const int ki = j & 127;
    const int k  = ki >> 6, i = ki & 63;
    const float* src = (k == 0) ? c : xg2;
    xl[b][ki] = src[((size_t)b * NN + n) * LL + i];
  }
  __syncthreads();

  // 512 outputs (b,o) -> 2 per thread
#pragma unroll
  for (int u = 0; u < 2; ++u) {
    const int idx = t + u * 256;    // idx = b*64 + o
    const int b = idx >> 6, o = idx & 63;
    float acc = 0.f;
#pragma unroll
    for (int d = 0; d < ADA; ++d) acc = fmaf(En[d], bp[d * LL + o], acc);
    for (int ki = 0; ki < 2 * LL; ++ki)
      acc = fmaf(xl[b][ki], Wn[ki * LL + o], acc);
    ct[((size_t)b * NN + n) * LL + o] = tanhf(acc);
  }
}

// ---------------------------------------------------------------------------
// K4: out_h[hb] = -softmax(theta theta^T / 4) @ ct_h[hb]   (flash-style)
// theta(hb,n,d) = state[b,n,h*16+d], ct_h likewise; hb = h*B + b
// grid: (N/16, HB); block: 32 (one wave, one 16-query tile)
// ---------------------------------------------------------------------------
__global__ void attn_kernel(const float* __restrict__ state,
                            const float* __restrict__ ct,
                            float* __restrict__ out) {
  const int qt = blockIdx.x;
  const int hb = blockIdx.y;
  const int b = hb % BB, h = hb / BB;
  const int lane = threadIdx.x;
  const int half = lane >> 4;
  const int lid  = lane & 15;
  const int q0 = qt * 16;
  const size_t base = (size_t)b * NN * LL + (size_t)h * DD;  // + n*LL + d

  __shared__ float pl[16][17];  // P tile (padded to dodge bank conflicts)

  // Preload Q fragments: step s covers K-dims (d) [s*4, s*4+4)
  v2f qf[4];
#pragma unroll
  for (int s = 0; s < 4; ++s) {
    const int kd = s * 4 + half * 2;
    const float* qp = state + base + (size_t)(q0 + lid) * LL + kd;
    qf[s].x = qp[0];
    qf[s].y = qp[1];
  }

  v8f oacc = {};
  float run_max[8], run_sum[8];
#pragma unroll
  for (int r = 0; r < 8; ++r) { run_max[r] = -INFINITY; run_sum[r] = 0.f; }

  for (int key0 = 0; key0 < NN; key0 += 16) {
    // S = Q K^T   (M=16 queries, N=16 keys, contraction over d=16)
    v8f sacc = {};
#pragma unroll
    for (int s = 0; s < 4; ++s) {
      const int kd = s * 4 + half * 2;
      const float* kp = state + base + (size_t)(key0 + lid) * LL + kd;
      v2f kf = { kp[0], kp[1] };
      sacc = wmma_f32_k4(qf[s], kf, sacc);
    }

    __syncthreads();  // protect pl from previous iteration's reads

    // online softmax per query row (row m = r + 8*half; 16 key cols across lanes)
#pragma unroll
    for (int r = 0; r < 8; ++r) {
      float sv = sacc[r] * 0.25f;  // 1/sqrt(16)
      float mx = sv;
      mx = fmaxf(mx, __shfl_xor(mx, 1));
      mx = fmaxf(mx, __shfl_xor(mx, 2));
      mx = fmaxf(mx, __shfl_xor(mx, 4));
      mx = fmaxf(mx, __shfl_xor(mx, 8));
      const float nm = fmaxf(run_max[r], mx);
      const float p = __expf(sv - nm);
      float ps = p;
      ps += __shfl_xor(ps, 1);
      ps += __shfl_xor(ps, 2);
      ps += __shfl_xor(ps, 4);
      ps += __shfl_xor(ps, 8);
      const float alpha = __expf(run_max[r] - nm);
      run_sum[r] = run_sum[r] * alpha + ps;
      run_max[r] = nm;
      oacc[r] *= alpha;
      pl[r + 8 * half][lid] = p;   // P in (q-row, key-col) order
    }
    __syncthreads();

    // O += P @ V   (contraction over 16 keys -> 4 wmma)
#pragma unroll
    for (int s = 0; s < 4; ++s) {
      const int kk = s * 4 + half * 2;              // key slice for this half
      v2f af = { pl[lid][kk], pl[lid][kk + 1] };    // A-frag: row=query(lid)
      const float* vp = ct + base + (size_t)(key0 + kk) * LL + lid;
      v2f bf = { vp[0], vp[LL] };                   // V[kk][dcol], V[kk+1][dcol]
      oacc = wmma_f32_k4(af, bf, oacc);
    }
  }

  // normalize, negate, write out[b, n, h*16 + d]
#pragma unroll
  for (int r = 0; r < 8; ++r) {
    const float o = -(oacc[r] / run_sum[r]);
    out[base + (size_t)(q0 + r + 8 * half) * LL + lid] = o;
  }
}

// ---------------------------------------------------------------------------
extern "C" void kernel_launch(void* const* d_in, const int* in_sizes, int n_in,
                              void* d_out, int out_size, void* d_ws, size_t ws_size,
                              hipStream_t stream) {
  // dict order: 0 c, 1 c_condition, 2 c_disease, 3 state, 4 emb_normal,
  //             5..12 dead query-path weights, 13 Wp_gcn, 14 bp_gcn
  const float* c     = (const float*)d_in[0];
  const float* state = (const float*)d_in[3];
  const float* E     = (const float*)d_in[4];
  const float* Wp    = (const float*)d_in[13];
  const float* bp    = (const float*)d_in[14];
  float* out = (float*)d_out;

  float* A   = (float*)d_ws;                 // [N,N]      16 MB
  float* xg2 = A + (size_t)NN * NN;          // [B,N,L]     4 MB
  float* ct  = xg2 + (size_t)BB * NN * LL;   // [B,N,L]     4 MB

  adj_kernel<<<NN, 256, 0, stream>>>(E, A);

  dim3 g2(NN / 64, BB);
  gemm_A_c<<<g2, 128, 0, stream>>>(A, c, xg2);

  gcn_kernel<<<NN, 256, 0, stream>>>(c, xg2, E, Wp, bp, ct);

  dim3 g4(NN / 16, HB);
  attn_kernel<<<g4, 32, 0, stream>>>(state, ct, out);
}